// Laplace3D_86620900425862
// MI455X (gfx1250) — compile-verified
//
#include <hip/hip_runtime.h>
#include <hip/hip_bf16.h>
#include <stdint.h>

// ---------------- problem constants (from reference) ----------------
constexpr int N_ = 4;     // batch
constexpr int D_ = 128;   // depth
constexpr int H_ = 256;   // height
constexpr int W_ = 256;   // width

// ---------------- tiling ----------------
constexpr int TX = 64;                 // tile width  (w)
constexpr int TY = 32;                 // tile height (h)
constexpr int LW = TX + 2;             // 66 (halo); odd-ish stride keeps LDS reads bank-conflict-free
constexpr int LH = TY + 2;             // 34 (halo)
constexpr int PLANE = LW * LH;         // 2244 floats per LDS plane
constexpr int PLANE1 = PLANE + 1;      // +1 sink cell per ring slot
constexpr int NTHREADS = 256;          // 8 waves (wave32)
constexpr int KLOADS = (PLANE + NTHREADS - 1) / NTHREADS;  // 9 async-load issues/wave/plane
constexpr int CHUNK = 32;              // z-planes per block

typedef __attribute__((address_space(1))) int GlobalInt;  // global src operand
typedef __attribute__((address_space(3))) int LdsInt;     // LDS dst operand
typedef float f32x4 __attribute__((ext_vector_type(4)));  // plain vector for NT store

#define HAS_ASYNC_LDS __has_builtin(__builtin_amdgcn_global_load_async_to_lds_b32)

__device__ __forceinline__ void async_copy_b32(const float* g, float* l) {
#if HAS_ASYNC_LDS
  // gfx1250: GLOBAL_LOAD_ASYNC_TO_LDS_B32 (tracked by ASYNCcnt)
  __builtin_amdgcn_global_load_async_to_lds_b32((GlobalInt*)g, (LdsInt*)l, 0, 0);
#else
  *l = *g;  // portable fallback: plain load + ds_store (tracked by DScnt)
#endif
}

template <int Ncnt>
__device__ __forceinline__ void wait_async() {
#if HAS_ASYNC_LDS
#if __has_builtin(__builtin_amdgcn_s_wait_asynccnt)
  __builtin_amdgcn_s_wait_asynccnt(Ncnt);
#else
  asm volatile("s_wait_asynccnt %0" ::"i"(Ncnt) : "memory");
#endif
#endif
}

__global__ __launch_bounds__(NTHREADS)
void Laplace3D_86620900425862_kernel(const float* __restrict__ x,
                                     float* __restrict__ out) {
  // 4-slot ring of halo planes, each with a trailing sink cell for masked
  // async lanes (~35.9 KB total).
  __shared__ float shm[4 * PLANE1];

  const int tid = threadIdx.x;
  const int tileW = blockIdx.x & 3;   // W_/TX = 4
  const int tileH = blockIdx.x >> 2;  // H_/TY = 8
  const int w0 = tileW * TX;
  const int h0 = tileH * TY;
  const int d0 = blockIdx.y * CHUNK;
  const int n  = blockIdx.z;
  const int dend  = d0 + CHUNK;
  const int pLast = dend;             // planes d0-1 .. dend are needed

  // ---- plane-invariant per-k async-copy operands (held in VGPRs) ----
  // goff[k]: element offset within one z-plane (0 when lane is masked)
  // eoff[k]: slot-relative LDS element offset (PLANE = sink when masked)
  int goff[KLOADS];
  int eoff[KLOADS];
#pragma unroll
  for (int k = 0; k < KLOADS; ++k) {
    const int e = tid + k * NTHREADS;
    const int r = e / LW, c = e - r * LW;
    const int gh = h0 - 1 + r, gw = w0 - 1 + c;
    const bool ok = (e < PLANE) & (gh >= 0) & (gh < H_) & (gw >= 0) & (gw < W_);
    goff[k] = ok ? (gh * W_ + gw) : 0;
    eoff[k] = ok ? e : PLANE;
  }

  // Zero the LDS cells whose (h,w) fall outside the volume, once, in all 4
  // slots. Async loads never target these cells, so they stay zero.
  for (int e = tid; e < PLANE; e += NTHREADS) {
    const int r = e / LW, c = e - r * LW;
    const int gh = h0 - 1 + r, gw = w0 - 1 + c;
    if (gh < 0 || gh >= H_ || gw < 0 || gw >= W_) {
      shm[0 * PLANE1 + e] = 0.f;
      shm[1 * PLANE1 + e] = 0.f;
      shm[2 * PLANE1 + e] = 0.f;
      shm[3 * PLANE1 + e] = 0.f;
    }
  }
  __syncthreads();

  // Exactly KLOADS full-EXEC async-load instructions per wave per plane, so
  // ASYNCcnt arithmetic is identical in every wave and s_wait_asynccnt<KLOADS>
  // is always safe.
  auto issue_plane_async = [&](int p) {  // 0 <= p < D_
    float* dst = &shm[(p & 3) * PLANE1];
    const float* gb = x + (size_t)(n * D_ + p) * (size_t)(H_ * W_);
#pragma unroll
    for (int k = 0; k < KLOADS; ++k) async_copy_b32(gb + goff[k], dst + eoff[k]);
  };
  auto zero_plane = [&](int p) {  // z-out-of-range plane -> zeros (DS stores)
    float* dst = &shm[(p & 3) * PLANE1];
    for (int e = tid; e < PLANE1; e += NTHREADS) dst[e] = 0.f;
  };
  auto load_plane = [&](int p) -> bool {  // true iff async loads were issued
    if (p >= 0 && p < D_) { issue_plane_async(p); return true; }
    zero_plane(p);
    return false;
  };

  // ---- preload planes d0-1, d0, d0+1 ----
  load_plane(d0 - 1);
  load_plane(d0);
  load_plane(d0 + 1);
  wait_async<0>();
  __syncthreads();

  const int tq = tid & 15;   // 16 groups of 4 consecutive w
  const int ty = tid >> 4;   // rows ty and ty+16
  const int c0 = tq * 4;

  for (int d = d0; d < dend; ++d) {
    // (1) prefetch plane d+2 into slot (d+2)&3 == (d-2)&3 (freed by the
    //     trailing barrier of the previous iteration)
    bool didAsync = false;
    const int pf = d + 2;
    if (pf <= pLast) didAsync = load_plane(pf);

    // (2) ensure plane d+1 (and older) are resident; leave only the new
    //     prefetch in flight so its DMA overlaps this plane's compute
    if (didAsync) wait_async<KLOADS>(); else wait_async<0>();
    __syncthreads();

    // (3) compute plane d: 7-point stencil from LDS, 2x float4 per thread
    const float* prv = &shm[((d - 1) & 3) * PLANE1];
    const float* cur = &shm[((d    ) & 3) * PLANE1];
    const float* nxt = &shm[((d + 1) & 3) * PLANE1];
#pragma unroll
    for (int rr = 0; rr < 2; ++rr) {
      const int y = ty + rr * 16;
      const int base = (y + 1) * LW + (c0 + 1);
      const float* cC = cur + base;   // center row
      const float* cU = cC - LW;      // h-1
      const float* cD = cC + LW;      // h+1
      const float* cP = prv + base;   // d-1
      const float* cN = nxt + base;   // d+1
      f32x4 o;
      o.x = cC[-1] + cC[1] + cU[0] + cD[0] + cP[0] + cN[0] - 6.0f * cC[0];
      o.y = cC[ 0] + cC[2] + cU[1] + cD[1] + cP[1] + cN[1] - 6.0f * cC[1];
      o.z = cC[ 1] + cC[3] + cU[2] + cD[2] + cP[2] + cN[2] - 6.0f * cC[2];
      o.w = cC[ 2] + cC[4] + cU[3] + cD[3] + cP[3] + cN[3] - 6.0f * cC[3];
      const size_t oidx =
          ((size_t)(n * D_ + d) * H_ + (size_t)(h0 + y)) * W_ + (w0 + c0);
      __builtin_nontemporal_store(o, reinterpret_cast<f32x4*>(out + oidx));
    }

    // (4) all reads of slot (d-1)&3 must finish before the next iteration's
    //     prefetch (plane d+3) overwrites it
    __syncthreads();
  }
}

extern "C" void kernel_launch(void* const* d_in, const int* in_sizes, int n_in,
                              void* d_out, int out_size, void* d_ws, size_t ws_size,
                              hipStream_t stream) {
  (void)in_sizes; (void)n_in; (void)out_size; (void)d_ws; (void)ws_size;
  const float* x = (const float*)d_in[0];
  float* out = (float*)d_out;
  dim3 grid((W_ / TX) * (H_ / TY), D_ / CHUNK, N_);  // (32, 4, 4)
  dim3 block(NTHREADS);
  hipLaunchKernelGGL(Laplace3D_86620900425862_kernel, grid, block, 0, stream, x, out);
}